// OverlapLoss_30657476558964
// MI455X (gfx1250) — compile-verified
//
#include <hip/hip_runtime.h>
#include <math.h>

typedef __attribute__((ext_vector_type(2))) float v2f;
typedef __attribute__((ext_vector_type(8))) float v8f;

#define ROW_TILES 4   // 16-row tiles per wave (64 rows); all sizes here are /64

// Single-instruction IEEE minimumNumber: avoids the compiler's conservative
// v_max_num canonicalize guards around llvm.minnum (the HW op already quiets
// NaNs per ISA ch.12), and keeps the accumulator opaque so it is not
// re-canonicalized every loop iteration.
__device__ __forceinline__ float min_num(float a, float b) {
    float r;
    asm("v_min_num_f32 %0, %1, %2" : "=v"(r) : "v"(a), "v"(b));
    return r;
}

// ---------------------------------------------------------------------------
// Pack points into float4 (x,y,z,|p|^2), optionally applying the rigid
// transform (row-major 4x4, p' = T[:3,:3] @ p + T[:3,3]).
// ---------------------------------------------------------------------------
__global__ void pack_points_kernel(const float* __restrict__ pts, int n,
                                   const float* __restrict__ T, int apply_t,
                                   float4* __restrict__ out) {
    int i = blockIdx.x * blockDim.x + threadIdx.x;
    if (i >= n) return;
    float x = pts[3 * i + 0];
    float y = pts[3 * i + 1];
    float z = pts[3 * i + 2];
    if (apply_t) {
        float nx = T[0] * x + T[1] * y + T[2]  * z + T[3];
        float ny = T[4] * x + T[5] * y + T[6]  * z + T[7];
        float nz = T[8] * x + T[9] * y + T[10] * z + T[11];
        x = nx; y = ny; z = nz;
    }
    out[i] = make_float4(x, y, z, x * x + y * y + z * z);
}

// ---------------------------------------------------------------------------
// Pre-pack the WMMA B fragments for a (already transformed) point set:
//   tile t, lane l<16 : (-2x, -2y) of point t*16+l     (K=0,1 rows of B)
//   tile t, lane l>=16: (-2z, |s|^2) of point t*16+l-16 (K=2,3 rows of B)
// Inner loop then needs only one b64 load per lane per tile, no VALU.
// ---------------------------------------------------------------------------
__global__ void pack_bfrag_kernel(const float4* __restrict__ P, int n,
                                  float2* __restrict__ Bfrag) {
    int i = blockIdx.x * blockDim.x + threadIdx.x;
    if (i >= n) return;
    float4 p = P[i];
    int t = i >> 4;
    int m = i & 15;
    Bfrag[t * 32 + m]      = make_float2(-2.0f * p.x, -2.0f * p.y);
    Bfrag[t * 32 + 16 + m] = make_float2(-2.0f * p.z, p.w);
}

// ---------------------------------------------------------------------------
// Initialize min buffers to +inf bit pattern (ws is poisoned by the harness).
// ---------------------------------------------------------------------------
__global__ void init_inf_kernel(unsigned int* __restrict__ p, int n) {
    int i = blockIdx.x * blockDim.x + threadIdx.x;
    if (i < n) p[i] = 0x7F800000u;  // +inf
}

// ---------------------------------------------------------------------------
// Core kernel: ROW_TILES 16-row tiles of A per wave; stream prepacked B
// fragments; V_WMMA_F32_16X16X4_F32 computes squared-distance tiles:
//   D = A x B + C, A[m] = (x,y,z,1), B[:,n] = (-2x,-2y,-2z,|s|^2), C = |r|^2
// One wave per block. gridDim.x = rows/(16*ROW_TILES), gridDim.y = chunks.
// Row minima merged with atomicMin on float bits (sqdist >= 0 so uint order
// == float order).
// ---------------------------------------------------------------------------
__global__ void __launch_bounds__(32)
min_sqdist_wmma_kernel(const float4* __restrict__ A,
                       const float2* __restrict__ Bfrag,  // [nTilesB*32]
                       int tilesB_per_chunk,
                       unsigned int* __restrict__ rowmin_bits) {
    const int lane = threadIdx.x & 31;
    const int half = lane >> 4;   // 0: K=0,1 / M half 0 ; 1: K=2,3 / half 1
    const int l16  = lane & 15;
    const int rowBase = blockIdx.x * (16 * ROW_TILES);

    const float INF = __uint_as_float(0x7F800000u);
    v2f afrag[ROW_TILES];
    v8f cfrag[ROW_TILES];
    v8f acc[ROW_TILES];
#pragma unroll
    for (int rt = 0; rt < ROW_TILES; ++rt) {
        // A fragment: lane<16 -> (x,y) of its row ; lane>=16 -> (z, 1)
        float4 pa = A[rowBase + rt * 16 + l16];
        afrag[rt].x = half ? pa.z : pa.x;
        afrag[rt].y = half ? 1.0f : pa.y;
        // C fragment: c[r] = |row (r + 8*half)|^2   (C/D layout)
#pragma unroll
        for (int r = 0; r < 8; ++r) {
            cfrag[rt][r] = A[rowBase + rt * 16 + 8 * half + r].w;
            acc[rt][r]   = INF;
        }
    }

    const int tile0 = blockIdx.y * tilesB_per_chunk;
    const float2* bp = Bfrag + (size_t)tile0 * 32 + lane;
    for (int t = 0; t < tilesB_per_chunk; ++t) {
        float2 b = bp[t * 32];        // global_load_b64, fragment-ready
        v2f bfrag;
        bfrag.x = b.x;
        bfrag.y = b.y;
#pragma unroll
        for (int rt = 0; rt < ROW_TILES; ++rt) {
            v8f d = __builtin_amdgcn_wmma_f32_16x16x4_f32(
                /*neg_a=*/false, afrag[rt], /*neg_b=*/false, bfrag,
                /*c_mod=*/(short)0, cfrag[rt],
                /*reuse_a=*/false, /*reuse_b=*/false);
            // Row identity of each (vgpr, lane-half) slot is tile-invariant:
            // elementwise min across tiles preserves per-row minima.
#pragma unroll
            for (int r = 0; r < 8; ++r) acc[rt][r] = min_num(acc[rt][r], d[r]);
        }
    }

    // Butterfly min across the 16 lanes of each half (N dimension), then one
    // atomic per row from lane 0 of each half.
#pragma unroll
    for (int rt = 0; rt < ROW_TILES; ++rt) {
#pragma unroll
        for (int r = 0; r < 8; ++r) {
            float v = acc[rt][r];
            v = min_num(v, __shfl_xor(v, 1, 32));
            v = min_num(v, __shfl_xor(v, 2, 32));
            v = min_num(v, __shfl_xor(v, 4, 32));
            v = min_num(v, __shfl_xor(v, 8, 32));
            if (l16 == 0)
                atomicMin(rowmin_bits + rowBase + rt * 16 + 8 * half + r,
                          __float_as_uint(v));
        }
    }
}

// ---------------------------------------------------------------------------
// Class-balanced BCE over concat([pred_src, pred_ref]) with gt from squared
// min distances vs thr^2. Single block, two passes (positive count, then
// weighted loss). Writes one scalar.
// ---------------------------------------------------------------------------
__global__ void bce_kernel(const float* __restrict__ pred_src, int n_src,
                           const float* __restrict__ pred_ref, int n_ref,
                           const unsigned int* __restrict__ mins_src,
                           const unsigned int* __restrict__ mins_ref,
                           float thr_sq, float* __restrict__ out_slot) {
    __shared__ float red[256];
    const int tid = threadIdx.x;
    const int n_total = n_src + n_ref;

    // Pass 1: count positives
    float pos = 0.0f;
    for (int i = tid; i < n_total; i += 256) {
        float sq = __uint_as_float(i < n_src ? mins_src[i] : mins_ref[i - n_src]);
        pos += (sq < thr_sq) ? 1.0f : 0.0f;
    }
    red[tid] = pos;
    __syncthreads();
    for (int s = 128; s > 0; s >>= 1) {
        if (tid < s) red[tid] += red[tid + s];
        __syncthreads();
    }
    const float w_neg = red[0] / (float)n_total;
    const float w_pos = 1.0f - w_neg;
    __syncthreads();

    // Pass 2: weighted BCE
    float acc = 0.0f;
    for (int i = tid; i < n_total; i += 256) {
        float sq = __uint_as_float(i < n_src ? mins_src[i] : mins_ref[i - n_src]);
        float pred = (i < n_src) ? pred_src[i] : pred_ref[i - n_src];
        bool  gt = (sq < thr_sq);
        float cl = gt ? (-logf(pred)) : (-log1pf(-pred));
        acc += (gt ? w_pos : w_neg) * cl;
    }
    red[tid] = acc;
    __syncthreads();
    for (int s = 128; s > 0; s >>= 1) {
        if (tid < s) red[tid] += red[tid + s];
        __syncthreads();
    }
    if (tid == 0) *out_slot = red[0] / (float)n_total;
}

// ---------------------------------------------------------------------------
extern "C" void kernel_launch(void* const* d_in, const int* in_sizes, int n_in,
                              void* d_out, int out_size, void* d_ws, size_t ws_size,
                              hipStream_t stream) {
    const float* ref_c = (const float*)d_in[0];
    const float* src_c = (const float*)d_in[1];
    const float* ref_f = (const float*)d_in[2];
    const float* src_f = (const float*)d_in[3];
    const float* T     = (const float*)d_in[4];
    const float* p2p_ref_scores = (const float*)d_in[5];
    const float* p2p_src_scores = (const float*)d_in[6];
    const float* n2p_ref_scores = (const float*)d_in[7];
    const float* n2p_src_scores = (const float*)d_in[8];
    float* out = (float*)d_out;

    const int Nc = in_sizes[0] / 3;   // 1024
    const int Mc = in_sizes[1] / 3;   // 1024
    const int Nf = in_sizes[2] / 3;   // 16384
    const int Mf = in_sizes[3] / 3;   // 16384

    // Workspace layout
    char* ws = (char*)d_ws;
    float4* ref_f4 = (float4*)ws;               ws += (size_t)Nf * sizeof(float4);
    float4* src_f4 = (float4*)ws;               ws += (size_t)Mf * sizeof(float4);
    float4* ref_c4 = (float4*)ws;               ws += (size_t)Nc * sizeof(float4);
    float4* src_c4 = (float4*)ws;               ws += (size_t)Mc * sizeof(float4);
    float2* bfrag_ref_f = (float2*)ws;          ws += (size_t)(Nf / 16) * 32 * sizeof(float2);
    float2* bfrag_src_f = (float2*)ws;          ws += (size_t)(Mf / 16) * 32 * sizeof(float2);
    unsigned int* min_all = (unsigned int*)ws;  // contiguous for one init pass
    unsigned int* min_ref_ff  = min_all;                 // [Nf] ref_gt mins
    unsigned int* min_src_ff  = min_ref_ff + Nf;         // [Mf] src_gt mins
    unsigned int* min_n2p_ref = min_src_ff + Mf;         // [Nc]
    unsigned int* min_n2p_src = min_n2p_ref + Nc;        // [Mc]
    const int n_min = Nf + Mf + Nc + Mc;

    // 1) Transform + pack (x,y,z,|p|^2)
    pack_points_kernel<<<(Nf + 255) / 256, 256, 0, stream>>>(ref_f, Nf, T, 0, ref_f4);
    pack_points_kernel<<<(Mf + 255) / 256, 256, 0, stream>>>(src_f, Mf, T, 1, src_f4);
    pack_points_kernel<<<(Nc + 255) / 256, 256, 0, stream>>>(ref_c, Nc, T, 0, ref_c4);
    pack_points_kernel<<<(Mc + 255) / 256, 256, 0, stream>>>(src_c, Mc, T, 1, src_c4);

    // 2) Pre-pack B fragments for the two streamed sets
    pack_bfrag_kernel<<<(Nf + 255) / 256, 256, 0, stream>>>(ref_f4, Nf, bfrag_ref_f);
    pack_bfrag_kernel<<<(Mf + 255) / 256, 256, 0, stream>>>(src_f4, Mf, bfrag_src_f);

    // 3) Init min buffers to +inf
    init_inf_kernel<<<(n_min + 255) / 256, 256, 0, stream>>>(min_all, n_min);

    // 4) WMMA min-squared-distance passes
    const int ROWS_PER_WAVE = 16 * ROW_TILES;   // 64
    const int CH_F = 32;   // chunks for 16384-col streams (fine grids)
    const int CH_C = 64;   // chunks for the small 1024-row grids
    {   // ref_f rows vs src_f_t cols  -> per-ref min (ref_gt)
        dim3 g(Nf / ROWS_PER_WAVE, CH_F);
        min_sqdist_wmma_kernel<<<g, 32, 0, stream>>>(ref_f4, bfrag_src_f,
                                                     (Mf / 16) / CH_F, min_ref_ff);
    }
    {   // src_f_t rows vs ref_f cols  -> per-src min (src_gt, transposed view)
        dim3 g(Mf / ROWS_PER_WAVE, CH_F);
        min_sqdist_wmma_kernel<<<g, 32, 0, stream>>>(src_f4, bfrag_ref_f,
                                                     (Nf / 16) / CH_F, min_src_ff);
    }
    {   // ref_c rows vs src_f_t cols  -> n2p ref mins
        dim3 g(Nc / ROWS_PER_WAVE, CH_C);
        min_sqdist_wmma_kernel<<<g, 32, 0, stream>>>(ref_c4, bfrag_src_f,
                                                     (Mf / 16) / CH_C, min_n2p_ref);
    }
    {   // src_c_t rows vs ref_f cols  -> n2p src mins
        dim3 g(Mc / ROWS_PER_WAVE, CH_C);
        min_sqdist_wmma_kernel<<<g, 32, 0, stream>>>(src_c4, bfrag_ref_f,
                                                     (Nf / 16) / CH_C, min_n2p_src);
    }

    // 5) Weighted BCE: out[0] = n2p_loss (thr 0.5), out[1] = p2p_loss (thr 0.1)
    bce_kernel<<<1, 256, 0, stream>>>(n2p_src_scores, Mc, n2p_ref_scores, Nc,
                                      min_n2p_src, min_n2p_ref, 0.25f, out + 0);
    bce_kernel<<<1, 256, 0, stream>>>(p2p_src_scores, Mf, p2p_ref_scores, Nf,
                                      min_src_ff, min_ref_ff, 0.01f, out + 1);
}